// GCNModel_90031104458821
// MI455X (gfx1250) — compile-verified
//
#include <hip/hip_runtime.h>

#define NN 50000      // nodes per type
#define EE 400000     // edges per relation
#define NREL 9

typedef __attribute__((ext_vector_type(2))) float v2f;
typedef __attribute__((ext_vector_type(8))) float v8f;

// REL = [(0,2),(1,2),(1,0),(0,0),(2,1),(2,0),(0,0),(2,2),(1,1)]
__device__ __constant__ int c_rel_dst[NREL] = {2, 2, 0, 0, 1, 0, 0, 2, 1};

// ---------------------------------------------------------------------------
// 1) Per-relation degree counts (float atomic +1.0 is exact for counts < 2^24)
// ---------------------------------------------------------------------------
__global__ void deg_count_kernel(const int* __restrict__ edges,
                                 float* __restrict__ deg_s,
                                 float* __restrict__ deg_d) {
    long long idx = (long long)blockIdx.x * blockDim.x + threadIdx.x;
    if (idx >= (long long)NREL * EE) return;
    int r = (int)(idx / EE);
    int e = (int)(idx % EE);
    int s = edges[(long long)r * 2 * EE + e];
    int d = edges[(long long)r * 2 * EE + EE + e];
    unsafeAtomicAdd(&deg_s[(long long)r * NN + s], 1.0f);
    unsafeAtomicAdd(&deg_d[(long long)r * NN + d], 1.0f);
}

// deg -> deg^-0.5 (0 when deg==0), in place over both norm buffers
__global__ void to_norm_kernel(float* __restrict__ v, long long n) {
    long long i = (long long)blockIdx.x * blockDim.x + threadIdx.x;
    if (i >= n) return;
    float d = v[i];
    v[i] = (d > 0.0f) ? __frsqrt_rn(d) : 0.0f;
}

// ---------------------------------------------------------------------------
// 2) Bias: bsum[t][f] = sum over relations r with dst(r)==t of b[r][f]
// ---------------------------------------------------------------------------
__global__ void bias_sum_kernel(const float* __restrict__ b, float* __restrict__ bsum) {
    int t = blockIdx.x;     // 3 blocks
    int f = threadIdx.x;    // 128 threads
    float acc = 0.0f;
#pragma unroll
    for (int r = 0; r < NREL; ++r)
        if (c_rel_dst[r] == t) acc += b[r * 128 + f];
    bsum[t * 128 + f] = acc;
}

__global__ void fill_bias_kernel(float* __restrict__ H, const float* __restrict__ bsum) {
    long long idx = (long long)blockIdx.x * blockDim.x + threadIdx.x;
    if (idx >= 3LL * NN * 128) return;
    int f = (int)(idx & 127);
    int t = (int)(idx / (128LL * NN));
    H[idx] = bsum[t * 128 + f];
}

// ---------------------------------------------------------------------------
// 3) WMMA FP32 GEMM: P[N,128] = (op(X)[N,128] * norm_s[row]) @ W[128,128]
//    op = identity (layer 1) or ReLU (layer 2). Uses V_WMMA_F32_16X16X4_F32.
//    Block = 256 threads = 8 waves; each wave owns a 16-row strip x all 128
//    output columns (8 f32 accumulator tiles).
//
//    W is staged through LDS in two 64-row halves using a FRAGMENT-INTERLEAVED
//    layout: Wlds[kp][n*2 + r] = W[2*kp + r][n], so a WMMA B fragment
//    {W[k][n], W[k+1][n]} is ONE aligned ds_load_b64 landing directly in an
//    even VGPR pair. Row stride 288 floats (=1152 B, bank shift 32) keeps the
//    two half-waves (kp vs kp+1) on disjoint bank sets.
//
//    Per k-step, ALL 8 B fragments are loaded into a dedicated fragment array
//    before the 8-WMMA chain, so the scheduler can issue the DS loads
//    back-to-back, wait once on dscnt, and stream the WMMAs stall-free.
// ---------------------------------------------------------------------------
template <int RELU>
__global__ __launch_bounds__(256) void gemm_proj_kernel(
    const float* __restrict__ X, const float* __restrict__ W,
    const float* __restrict__ norms, float* __restrict__ P, int nrows) {
    __shared__ float Wlds[32][288];   // 32 k-pairs x 128 col-pairs (+pad) = 36 KB

    const int wave = threadIdx.x >> 5;
    const int lane = threadIdx.x & 31;
    const int m0   = blockIdx.x * 128 + wave * 16;
    const int col  = lane & 15;       // A row / B,D column within tile
    const int hi   = lane >> 4;       // half-wave selects K pair / D row half
    const bool act = (m0 < nrows);

    v8f acc[8] = {};

    const int row = m0 + col;         // this lane's A-matrix source row
    float s = 0.0f;
    if (act) s = norms[row];

    for (int kh = 0; kh < 2; ++kh) {
        // Stage W rows [kh*64, kh*64+64) into interleaved LDS layout.
        // 32 kp x 64 col-pairs = 2048 float4 stores; 8 per thread.
        for (int i = threadIdx.x; i < 32 * 64; i += 256) {
            const int kp = i >> 6;            // k-pair 0..31
            const int c  = (i & 63) << 1;     // even column 0..126
            const int r0 = kh * 64 + kp * 2;
            const float2 w0 = *(const float2*)(W + (size_t)r0 * 128 + c);
            const float2 w1 = *(const float2*)(W + (size_t)(r0 + 1) * 128 + c);
            float4 pk;
            pk.x = w0.x; pk.y = w1.x; pk.z = w0.y; pk.w = w1.y;
            *(float4*)(&Wlds[kp][c * 2]) = pk;
        }
        if (kh == 0) {
            // Pull the second half of W toward the caches while we compute.
            __builtin_prefetch(W + 64 * 128 + threadIdx.x * 32, 0, 3);
        }
        __syncthreads();

        if (act) {
#pragma unroll 4
            for (int ks = 0; ks < 16; ++ks) {
                const int k  = kh * 64 + ks * 4 + 2 * hi;   // this lane's K base
                // A fragment: lanes 0-15 hold K={4ks,4ks+1}, lanes 16-31 K={+2,+3}
                float2 a2 = *(const float2*)(X + (size_t)row * 128 + k);
                float ax = a2.x, ay = a2.y;
                if (RELU) { ax = fmaxf(ax, 0.0f); ay = fmaxf(ay, 0.0f); }
                v2f a;
                a.x = ax * s;
                a.y = ay * s;

                // Load all 8 B fragments first (distinct registers), then
                // run the 8-WMMA chain without per-pair dscnt stalls.
                const float* brow = &Wlds[ks * 2 + hi][0];
                v2f bfrag[8];
#pragma unroll
                for (int nt = 0; nt < 8; ++nt)
                    bfrag[nt] = *(const v2f*)(brow + (nt * 16 + col) * 2);
#pragma unroll
                for (int nt = 0; nt < 8; ++nt) {
                    acc[nt] = __builtin_amdgcn_wmma_f32_16x16x4_f32(
                        /*neg_a=*/false, a, /*neg_b=*/false, bfrag[nt],
                        /*c_mod=*/(short)0, acc[nt],
                        /*reuse_a=*/false, /*reuse_b=*/false);
                }
            }
        }
        __syncthreads();
    }

    if (act) {
        // D layout: VGPR v -> row m0 + v + 8*hi, column nt*16 + (lane&15)
#pragma unroll
        for (int nt = 0; nt < 8; ++nt) {
#pragma unroll
            for (int v = 0; v < 8; ++v) {
                P[(size_t)(m0 + v + 8 * hi) * 128 + nt * 16 + col] = acc[nt][v];
            }
        }
    }
}

// ---------------------------------------------------------------------------
// 4) Edge scatter: H[dst] += norm_d[dst] * P[src]  (one wave per edge,
//    each lane handles 4 consecutive features; native f32 atomics)
// ---------------------------------------------------------------------------
__global__ __launch_bounds__(256) void scatter_kernel(
    const float* __restrict__ P, const int* __restrict__ src,
    const int* __restrict__ dst, const float* __restrict__ normd,
    float* __restrict__ H) {
    const int wave = threadIdx.x >> 5;
    const int lane = threadIdx.x & 31;
    const long long e = (long long)blockIdx.x * 8 + wave;
    if (e >= EE) return;
    const int s = src[e];
    const int d = dst[e];
    const float nd = normd[d];
    const float4 p4 = *(const float4*)(P + (size_t)s * 128 + lane * 4);
    float* out = H + (size_t)d * 128 + lane * 4;
    unsafeAtomicAdd(out + 0, p4.x * nd);
    unsafeAtomicAdd(out + 1, p4.y * nd);
    unsafeAtomicAdd(out + 2, p4.z * nd);
    unsafeAtomicAdd(out + 3, p4.w * nd);
}

// ---------------------------------------------------------------------------
// 5) Classifier: out[row][c] = relu(H2[row]) @ Wc[:,c] + bc[c]   (128 -> 16)
// ---------------------------------------------------------------------------
__global__ void classifier_kernel(const float* __restrict__ H2,
                                  const float* __restrict__ Wc,
                                  const float* __restrict__ bc,
                                  float* __restrict__ out) {
    long long idx = (long long)blockIdx.x * blockDim.x + threadIdx.x;
    if (idx >= 3LL * NN * 16) return;
    const int c = (int)(idx & 15);
    const long long row = idx >> 4;
    const float* h = H2 + row * 128;
    float acc = bc[c];
#pragma unroll 8
    for (int k = 0; k < 128; ++k)
        acc = fmaf(fmaxf(h[k], 0.0f), Wc[k * 16 + c], acc);
    out[idx] = acc;
}

// ---------------------------------------------------------------------------
extern "C" void kernel_launch(void* const* d_in, const int* in_sizes, int n_in,
                              void* d_out, int out_size, void* d_ws, size_t ws_size,
                              hipStream_t stream) {
    (void)in_sizes; (void)n_in; (void)out_size; (void)ws_size;

    const float* X[3] = {(const float*)d_in[0], (const float*)d_in[1], (const float*)d_in[2]};
    const float* W1 = (const float*)d_in[3];
    const float* b1 = (const float*)d_in[4];
    const float* W2 = (const float*)d_in[5];
    const float* b2 = (const float*)d_in[6];
    const float* Wc = (const float*)d_in[7];
    const float* bc = (const float*)d_in[8];
    const int* edges = (const int*)d_in[9];
    float* out = (float*)d_out;

    static const int rsrc[NREL] = {0, 1, 1, 0, 2, 2, 0, 2, 1};
    static const int rdst[NREL] = {2, 2, 0, 0, 1, 0, 0, 2, 1};

    // Workspace layout (floats): ~183 MB total
    float* ws     = (float*)d_ws;
    float* norm_s = ws;                          // [9*NN]
    float* norm_d = norm_s + 9LL * NN;           // [9*NN]
    float* bsum   = norm_d + 9LL * NN;           // [3*128]
    float* H1     = bsum + 3 * 128;              // [3*NN*128]
    float* H2     = H1 + 3LL * NN * 128;         // [3*NN*128]
    float* P      = H2 + 3LL * NN * 128;         // [NN*128]

    // --- norms (shared by both layers: same edges) ---
    hipMemsetAsync(norm_s, 0, sizeof(float) * 18LL * NN, stream);
    {
        long long t = (long long)NREL * EE;
        deg_count_kernel<<<(unsigned)((t + 255) / 256), 256, 0, stream>>>(edges, norm_s, norm_d);
    }
    to_norm_kernel<<<(18LL * NN + 255) / 256, 256, 0, stream>>>(norm_s, 18LL * NN);

    const unsigned gemm_grid  = (NN + 127) / 128;
    const unsigned scat_grid  = (EE + 7) / 8;
    const unsigned fill_grid  = (unsigned)((3LL * NN * 128 + 255) / 256);

    // --- layer 1: H1 = sum_r scatter(norm_d * ((X[s]*norm_s) @ W1[r])) + bias ---
    bias_sum_kernel<<<3, 128, 0, stream>>>(b1, bsum);
    fill_bias_kernel<<<fill_grid, 256, 0, stream>>>(H1, bsum);
    for (int r = 0; r < NREL; ++r) {
        gemm_proj_kernel<0><<<gemm_grid, 256, 0, stream>>>(
            X[rsrc[r]], W1 + (size_t)r * 128 * 128, norm_s + (size_t)r * NN, P, NN);
        scatter_kernel<<<scat_grid, 256, 0, stream>>>(
            P, edges + (size_t)r * 2 * EE, edges + (size_t)r * 2 * EE + EE,
            norm_d + (size_t)r * NN, H1 + (size_t)rdst[r] * NN * 128);
    }

    // --- layer 2: inputs relu(H1), fused into GEMM A-load ---
    bias_sum_kernel<<<3, 128, 0, stream>>>(b2, bsum);
    fill_bias_kernel<<<fill_grid, 256, 0, stream>>>(H2, bsum);
    for (int r = 0; r < NREL; ++r) {
        gemm_proj_kernel<1><<<gemm_grid, 256, 0, stream>>>(
            H1 + (size_t)rsrc[r] * NN * 128, W2 + (size_t)r * 128 * 128,
            norm_s + (size_t)r * NN, P, NN);
        scatter_kernel<<<scat_grid, 256, 0, stream>>>(
            P, edges + (size_t)r * 2 * EE, edges + (size_t)r * 2 * EE + EE,
            norm_d + (size_t)r * NN, H2 + (size_t)rdst[r] * NN * 128);
    }

    // --- classifier with fused ReLU ---
    classifier_kernel<<<(unsigned)((3LL * NN * 16 + 255) / 256), 256, 0, stream>>>(
        H2, Wc, bc, out);
}